// GIN_57337813402032
// MI455X (gfx1250) — compile-verified
//
#include <hip/hip_runtime.h>

// GIN 2-layer forward for MI455X (gfx1250, wave32).
//   - aggregation: L2-resident gather + global_atomic_add_f32; lane-interleaved
//     layout so every load/atomic instruction covers 128 contiguous bytes
//   - GEMMs: fp32 WMMA (v_wmma_f32_16x16x4_f32), one 16-row strip per wave,
//     8 accumulator tiles covering all 128 output columns
//   - BN: coalesced column reduction -> fused scale/shift -> elementwise BN+ReLU
// Workspace need: 3 * 50000*128 floats + 512 floats  (~77 MB).

typedef float v2f __attribute__((ext_vector_type(2)));
typedef float v8f __attribute__((ext_vector_type(8)));

#define GIN_D 128

// ---------------- aggregation: out[dst] += h[src], one wave per edge ----------
// Lane l handles columns l, l+32, l+64, l+96: each global_load_b32 and each
// global_atomic_add_f32 instruction touches 32 consecutive floats (2 cachelines).
__global__ __launch_bounds__(256) void gin_edges_kernel(
    const float* __restrict__ h, const int* __restrict__ src,
    const int* __restrict__ dst, float* __restrict__ out, int nedges)
{
    unsigned gid = blockIdx.x * blockDim.x + threadIdx.x;
    int e = (int)(gid >> 5);              // wave id == edge id
    if (e >= nedges) return;              // wave-uniform exit
    int lane = threadIdx.x & 31;          // wave32
    int s = src[e];
    int d = dst[e];
    const float* p = h   + (size_t)s * GIN_D + lane;
    float*       o = out + (size_t)d * GIN_D + lane;
    float v0 = p[0];
    float v1 = p[32];
    float v2 = p[64];
    float v3 = p[96];
    atomicAdd(o + 0,  v0);
    atomicAdd(o + 32, v1);
    atomicAdd(o + 64, v2);
    atomicAdd(o + 96, v3);
}

// ---------------- GEMM: out = X @ W^T + bias (optional ReLU) ------------------
// X:[nrows,128] row-major, W:[128,128] row-major (W[out][in]).
// One wave computes rows [m0, m0+16) x all 128 cols via 8 WMMA accumulators.
template <bool RELU>
__global__ __launch_bounds__(256) void gemm_wmma_kernel(
    const float* __restrict__ X, const float* __restrict__ W,
    const float* __restrict__ bias, float* __restrict__ out, int nrows)
{
    unsigned gid = blockIdx.x * blockDim.x + threadIdx.x;
    int wid  = (int)(gid >> 5);
    int lane = threadIdx.x & 31;
    int m0 = wid * 16;
    if (m0 >= nrows) return;              // wave-uniform: EXEC stays all-ones
    int lr = lane & 15;                   // 0..15
    int hi = lane >> 4;                   // 0: K=k0..k0+1, 1: K=k0+2..k0+3

    // A frag: X[(m0+lr)][k0 + 2*hi .. +1]   (contiguous float2)
    // B frag: B[k][n] = W[n][k] -> W[(n0+lr)][k0 + 2*hi .. +1] (contiguous float2)
    const float* xp = X + (size_t)(m0 + lr) * GIN_D + 2 * hi;
    const float* wp = W + (size_t)lr * GIN_D + 2 * hi;

    v8f acc[8] = {};
#pragma unroll 4
    for (int kc = 0; kc < 32; ++kc) {
        int k0 = kc * 4;
        v2f a = *(const v2f*)(xp + k0);
#pragma unroll
        for (int nt = 0; nt < 8; ++nt) {
            v2f b = *(const v2f*)(wp + nt * (16 * GIN_D) + k0);
            acc[nt] = __builtin_amdgcn_wmma_f32_16x16x4_f32(
                false, a, false, b, (short)0, acc[nt], false, false);
        }
    }
    // C/D layout: VGPR r -> row m0 + r + 8*hi, col = nt*16 + lr
#pragma unroll
    for (int nt = 0; nt < 8; ++nt) {
        int col = nt * 16 + lr;
        float bv = bias[col];
#pragma unroll
        for (int r = 0; r < 8; ++r) {
            int row = m0 + 8 * hi + r;
            float v = acc[nt][r] + bv;
            if (RELU) v = fmaxf(v, 0.0f);
            out[(size_t)row * GIN_D + col] = v;
        }
    }
}

// ---------------- BN: column sums / sums-of-squares ---------------------------
__global__ __launch_bounds__(128) void bn_stats_kernel(
    const float* __restrict__ x, float* __restrict__ stats, int nrows)
{
    int col = threadIdx.x;                               // 128 threads = 4 waves
    int rows_per = (nrows + gridDim.x - 1) / gridDim.x;
    int r0 = blockIdx.x * rows_per;
    int r1 = min(r0 + rows_per, nrows);
    float s = 0.0f, sq = 0.0f;
    for (int r = r0; r < r1; ++r) {                      // coalesced: lane==col
        float v = x[(size_t)r * GIN_D + col];
        s  += v;
        sq += v * v;
    }
    atomicAdd(&stats[col], s);
    atomicAdd(&stats[GIN_D + col], sq);
}

__global__ __launch_bounds__(256) void bn_zero_kernel(float* __restrict__ stats)
{
    stats[threadIdx.x] = 0.0f;                           // 256 floats: sum+sumsq
}

__global__ __launch_bounds__(128) void bn_finalize_kernel(
    float* __restrict__ stats, const float* __restrict__ gamma,
    const float* __restrict__ beta, float inv_n)
{
    int c = threadIdx.x;
    float mean  = stats[c] * inv_n;
    float var   = stats[GIN_D + c] * inv_n - mean * mean;
    float scale = gamma[c] * rsqrtf(var + 1e-5f);
    stats[2 * GIN_D + c] = scale;                        // fused scale
    stats[3 * GIN_D + c] = beta[c] - mean * scale;       // fused shift
}

__global__ __launch_bounds__(256) void bn_apply_relu_kernel(
    const float* __restrict__ x, const float* __restrict__ stats,
    float* __restrict__ out, int n4)
{
    int i = blockIdx.x * blockDim.x + threadIdx.x;
    if (i >= n4) return;
    int c = (i & 31) * 4;                                // 32 float4 per 128-col row
    float4 v  = ((const float4*)x)[i];
    float4 sc = *(const float4*)(stats + 2 * GIN_D + c);
    float4 sh = *(const float4*)(stats + 3 * GIN_D + c);
    float4 o;
    o.x = fmaxf(v.x * sc.x + sh.x, 0.0f);
    o.y = fmaxf(v.y * sc.y + sh.y, 0.0f);
    o.z = fmaxf(v.z * sc.z + sh.z, 0.0f);
    o.w = fmaxf(v.w * sc.w + sh.w, 0.0f);
    ((float4*)out)[i] = o;
}

// -----------------------------------------------------------------------------
extern "C" void kernel_launch(void* const* d_in, const int* in_sizes, int n_in,
                              void* d_out, int out_size, void* d_ws, size_t ws_size,
                              hipStream_t stream)
{
    const float* h   = (const float*)d_in[0];
    const int*   src = (const int*)  d_in[1];
    const int*   dst = (const int*)  d_in[2];
    const float* W1  = (const float*)d_in[3];
    const float* b1  = (const float*)d_in[4];
    const float* g1  = (const float*)d_in[5];
    const float* be1 = (const float*)d_in[6];
    const float* W2a = (const float*)d_in[7];
    const float* b2a = (const float*)d_in[8];
    const float* g2  = (const float*)d_in[9];
    const float* be2 = (const float*)d_in[10];
    const float* W2b = (const float*)d_in[11];
    const float* b2b = (const float*)d_in[12];
    float* out = (float*)d_out;

    const int N  = in_sizes[0] / GIN_D;          // 50000
    const int E  = in_sizes[1];                  // 800000
    const size_t NN = (size_t)N * GIN_D;
    const size_t NNbytes = NN * sizeof(float);

    float* B0    = (float*)d_ws;                 // aggregate buffer
    float* B1    = B0 + NN;                      // pre-BN linear output
    float* B2    = B1 + NN;                      // post BN+ReLU activations
    float* stats = B2 + NN;                      // 512 floats: sum/sumsq/scale/shift

    const int edgeBlocks  = (E * 32 + 255) / 256;
    const int gemmThreads = (N / 16) * 32;       // one wave per 16-row strip
    const int gemmBlocks  = (gemmThreads + 255) / 256;
    const int n4          = (int)(NN / 4);
    const int applyBlocks = (n4 + 255) / 256;
    const float inv_n     = 1.0f / (float)N;

    // ---- conv1: aggregate -> linear -> BN -> ReLU ----
    hipMemcpyAsync(B0, h, NNbytes, hipMemcpyDeviceToDevice, stream);   // self term
    gin_edges_kernel<<<edgeBlocks, 256, 0, stream>>>(h, src, dst, B0, E);
    gemm_wmma_kernel<false><<<gemmBlocks, 256, 0, stream>>>(B0, W1, b1, B1, N);
    bn_zero_kernel<<<1, 256, 0, stream>>>(stats);
    bn_stats_kernel<<<256, 128, 0, stream>>>(B1, stats, N);
    bn_finalize_kernel<<<1, 128, 0, stream>>>(stats, g1, be1, inv_n);
    bn_apply_relu_kernel<<<applyBlocks, 256, 0, stream>>>(B1, stats, B2, n4);

    // ---- conv2: aggregate -> linear -> BN -> ReLU -> linear -> ReLU ----
    hipMemcpyAsync(B0, B2, NNbytes, hipMemcpyDeviceToDevice, stream);  // self term
    gin_edges_kernel<<<edgeBlocks, 256, 0, stream>>>(B2, src, dst, B0, E);
    gemm_wmma_kernel<false><<<gemmBlocks, 256, 0, stream>>>(B0, W2a, b2a, B1, N);
    bn_zero_kernel<<<1, 256, 0, stream>>>(stats);
    bn_stats_kernel<<<256, 128, 0, stream>>>(B1, stats, N);
    bn_finalize_kernel<<<1, 128, 0, stream>>>(stats, g2, be2, inv_n);
    bn_apply_relu_kernel<<<applyBlocks, 256, 0, stream>>>(B1, stats, B2, n4);
    gemm_wmma_kernel<true><<<gemmBlocks, 256, 0, stream>>>(B2, W2b, b2b, out, N);
}